// TopoAELoss_27066883900124
// MI455X (gfx1250) — compile-verified
//
#include <hip/hip_runtime.h>

// TopoAE loss: two Prim MSTs over precomputed 4096x4096 distance matrices,
// loss fused into the MST selection step. Latency-bound serial algorithm:
// - per-node state (dist/parent) register-resident, one owner thread each
// - argmin: packed u64 wave32 shuffle reduce + single ds_min_u64 per wave
// - ONE barrier per iteration (3-slot rotating LDS min-slot, race-free)
// - read-once rows -> non-temporal b128 loads; both matrices warmed into L2
//   (128 MB < 192 MB L2) via global_prefetch_b8 in concurrent warm blocks.

#define NN 4096
#define T 512                 // threads per MST block (16 waves)
#define EPT (NN / T)          // 8 elements owned per thread
#define NWAVES (T / 32)       // 16
#define WARMB 16              // L2-warming blocks (8 per matrix)

typedef float v4f __attribute__((ext_vector_type(4)));

__device__ __forceinline__ unsigned long long shfl_xor_u64(unsigned long long v, int mask) {
    unsigned lo = (unsigned)(v & 0xffffffffull);
    unsigned hi = (unsigned)(v >> 32);
    lo = (unsigned)__shfl_xor((int)lo, mask, 32);
    hi = (unsigned)__shfl_xor((int)hi, mask, 32);
    return ((unsigned long long)hi << 32) | (unsigned long long)lo;
}

// Pack (distance, index) so u64-min == (min value, then min index).
// Distances are >= 0; sign-bit-set means "in tree" -> map to +inf.
__device__ __forceinline__ unsigned long long pack_key(float d, int idx) {
    unsigned b = __float_as_uint(d);
    b = (b & 0x80000000u) ? 0x7F800000u : b;
    return ((unsigned long long)b << 32) | (unsigned)idx;
}

__global__ void __launch_bounds__(T)
topoae_mst_kernel(const float* __restrict__ D1, const float* __restrict__ D2,
                  float* __restrict__ out) {
    const int b = blockIdx.x;
    const int tid = threadIdx.x;

    // ---------------- L2 warming blocks (b >= 2) ----------------
    if (b >= 2) {
        // Pull both matrices into the 192MB L2 (prefetch climbs the hierarchy),
        // so the MST blocks' serially-dependent row reads hit L2, not HBM.
        const size_t bytes = (size_t)NN * NN * sizeof(float);   // 64 MB each
        const int wb = b - 2;
        const int half = WARMB / 2;
        const unsigned char* base =
            (wb < half) ? (const unsigned char*)D1 : (const unsigned char*)D2;
        const int wslice = (wb < half) ? wb : wb - half;
        const size_t slice = bytes / half;                       // 8 MB per block
        const unsigned char* p0 = base + (size_t)wslice * slice;
        for (size_t off = (size_t)tid * 128; off < slice; off += (size_t)T * 128)
            __builtin_prefetch(p0 + off, 0, 3);                  // global_prefetch_b8
        return;
    }

    // ---------------- MST blocks (b == 0: D1-tree, b == 1: D2-tree) ----------------
    // 3 rotating argmin slots: atomics(it) -> slot[it%3]; reset(it) -> slot[(it+1)%3].
    // Reads of slot[(it+1)%3] were in epoch it-1 (sep. by barrier it-1); its next
    // atomics are in epoch it+1 (sep. by barrier it)  => single barrier is race-free.
    __shared__ unsigned long long slot[3];
    __shared__ float lossRed[NWAVES];

    const float* __restrict__ Dm = (b == 0) ? D1 : D2;   // tree built on this metric
    const float* __restrict__ Do = (b == 0) ? D2 : D1;   // compared metric

    const int lane = tid & 31;
    const int wid  = tid >> 5;
    const int i0   = tid * EPT;

    // Per-node state lives entirely in the owner thread's registers.
    float dist[EPT];
    int   par[EPT];
#pragma unroll
    for (int e = 0; e < EPT; ++e) {
        dist[e] = Dm[i0 + e];        // dist0 = D[0]
        par[e]  = 0;
    }
    if (tid == 0) {
        dist[0] = -__builtin_inff(); // node 0 in tree (masked forever)
        slot[0] = ~0ull; slot[1] = ~0ull; slot[2] = ~0ull;
    }
    __syncthreads();

    float loss = 0.0f;

    for (int it = 0; it < NN - 1; ++it) {
        const int cur = it % 3;
        // ---- argmin over all non-tree nodes (packed u64, first-index tiebreak) ----
        unsigned long long m = ~0ull;
#pragma unroll
        for (int e = 0; e < EPT; ++e) {
            unsigned long long k = pack_key(dist[e], i0 + e);
            m = (k < m) ? k : m;
        }
#pragma unroll
        for (int off = 16; off > 0; off >>= 1) {
            unsigned long long o = shfl_xor_u64(m, off);
            m = (o < m) ? o : m;
        }
        if (lane == 0) atomicMin(&slot[cur], m);           // ds_min_u64 (16 waves)
        if (tid == 0)  slot[(it + 1) % 3] = ~0ull;         // reset for next iter
        __syncthreads();                                   // the ONLY barrier

        const unsigned long long best = slot[cur];
        const int   j  = (int)(best & 0xffffffffull);
        const float dj = __uint_as_float((unsigned)(best >> 32));

        // ---- owner of j: fuse the loss term, mark j as in-tree ----
        if (i0 == (j & ~(EPT - 1))) {
            const int e = j - i0;
            const int p = par[e];
            const float other =
                __builtin_nontemporal_load(&Do[(size_t)p * NN + j]);
            const float diff = dj - other;                 // dj == Dm[p][j]
            loss += diff * diff;
            dist[e] = -__builtin_inff();
        }

        // ---- relax with row j (read-once: non-temporal b128 loads) ----
        const v4f* __restrict__ row = (const v4f*)(Dm + (size_t)j * NN + i0);
#pragma unroll
        for (int q = 0; q < EPT / 4; ++q) {
            v4f r = __builtin_nontemporal_load(&row[q]);
#pragma unroll
            for (int s = 0; s < 4; ++s) {
                const int e = q * 4 + s;
                const float rv = r[s];
                if (rv < dist[e]) { dist[e] = rv; par[e] = j; }  // strict <, as ref
            }
        }
    }

    // ---- block-wide loss reduction, single atomic per block ----
#pragma unroll
    for (int off = 16; off > 0; off >>= 1)
        loss += __shfl_xor(loss, off, 32);
    if (lane == 0) lossRed[wid] = loss;
    __syncthreads();
    if (tid == 0) {
        float s = 0.0f;
#pragma unroll
        for (int w = 0; w < NWAVES; ++w) s += lossRed[w];
        atomicAdd(out, s);
    }
}

__global__ void topoae_zero_kernel(float* __restrict__ out) {
    if (threadIdx.x == 0 && blockIdx.x == 0) out[0] = 0.0f;
}

extern "C" void kernel_launch(void* const* d_in, const int* in_sizes, int n_in,
                              void* d_out, int out_size, void* d_ws, size_t ws_size,
                              hipStream_t stream) {
    (void)in_sizes; (void)n_in; (void)out_size; (void)d_ws; (void)ws_size;
    const float* D1 = (const float*)d_in[0];   // input_distances  (4096*4096 f32)
    const float* D2 = (const float*)d_in[1];   // latent_distances (4096*4096 f32)
    float* out = (float*)d_out;

    hipLaunchKernelGGL(topoae_zero_kernel, dim3(1), dim3(1), 0, stream, out);
    hipLaunchKernelGGL(topoae_mst_kernel, dim3(2 + WARMB), dim3(T), 0, stream,
                       D1, D2, out);
}